// ConvCaps_23510650978795
// MI455X (gfx1250) — compile-verified
//
#include <hip/hip_runtime.h>
#include <math.h>

// ConvCaps (Bc=32, Cc=32, K=3, stride=2, pad=1, P=4) + dynamic routing (3 iters)
// for MI455X / gfx1250 (wave32, WMMA).
//
// Shapes: x (4,32,16,16,16) f32, a (4,32,16,16) f32 [UNUSED by outputs],
//         W (288,32,4,4) f32.  Outputs: p_out (4,32,16,8,8) + a_out (4,32,8,8).
//
// Plan (roofline): total math < 1.5 GFLOP; x+W = 2.6 MB -> fully L2 resident.
// Do NOT materialize the 151 MB voting tensor. Kernel 1 computes the voting
// einsum with V_WMMA_F32_16X16X4_F32 (K=4 == P, fp32-exact) just to produce
// the per-(b,C,s) max/min pose norms over B (64 KB workspace, uint atomics).
// Kernel 2 does the routing per spatial position, recomputing u on the fly
// from L2-resident x/W (cheap VALU 4x4x4 products).

#define EPS_F 1e-5f

typedef __attribute__((ext_vector_type(2))) float v2f;
typedef __attribute__((ext_vector_type(8))) float v8f;

// ---------------------------------------------------------------- init minmax
__global__ void init_minmax_kernel(unsigned int* __restrict__ mx,
                                   unsigned int* __restrict__ mn) {
    int g = blockIdx.x * blockDim.x + threadIdx.x;   // 8192 threads exactly
    mx[g] = 0u;           // 0.0f  (norms are > 0)
    mn[g] = 0x7f7fffffu;  // FLT_MAX
}

// ------------------------------------------------------- voting + norm minmax
// One block per (b, B): 4*288 = 1152 blocks, 256 threads (8 wave32).
// Per block: D[(C*4+k), (i*64+s)] = sum_j A[(C,k),j] * Bm[j,(i,s)]
//   A[(C*4+k), j] = W[B][C][j][k]          (128 x 4)
//   Bm[j, i*64+s] = x_patch[b,B,i*4+j,s]   (4 x 256)
// Each wave owns one 16-row M tile (4 C values) and loops 16 N tiles.
__global__ __launch_bounds__(256)
void vote_norm_kernel(const float* __restrict__ x,
                      const float* __restrict__ W,
                      unsigned int* __restrict__ mx,
                      unsigned int* __restrict__ mn) {
    __shared__ float Bmat[4 * 256];   // [j][i*64+s]
    __shared__ float Amat[128 * 4];   // [(C*4+k)][j]

    const int tid = threadIdx.x;
    const int blk = blockIdx.x;
    const int b     = blk / 288;
    const int B     = blk % 288;
    const int Borig = B / 9;
    const int kh    = (B % 9) / 3;
    const int kw    = B % 3;

    // ---- stage x patch: Bmat[j*256 + i*64 + s]
    {
        const int i  = tid >> 6;
        const int s  = tid & 63;
        const int oh = s >> 3, ow = s & 7;
        const int ih = oh * 2 - 1 + kh;
        const int iw = ow * 2 - 1 + kw;
        const bool ok = (ih >= 0) & (ih < 16) & (iw >= 0) & (iw < 16);
        const float* xb = x + (size_t)((b * 32 + Borig) * 16) * 256 + ih * 16 + iw;
#pragma unroll
        for (int j = 0; j < 4; ++j)
            Bmat[j * 256 + tid] = ok ? xb[(i * 4 + j) * 256] : 0.0f;
    }
    // ---- stage weights: Amat[(C*4+k)*4 + j] = W[B*512 + C*16 + j*4 + k]
    {
        const float* Wb = W + (size_t)B * 512;
#pragma unroll
        for (int h = 0; h < 2; ++h) {
            int f = tid + h * 256;
            int C = f >> 4;
            int j = (f >> 2) & 3;
            int k = f & 3;
            Amat[((C * 4 + k) << 2) | j] = Wb[f];
        }
    }
    __syncthreads();

    const int wave  = tid >> 5;
    const int lane  = tid & 31;
    const int lhalf = lane >> 4;   // 0/1
    const int l15   = lane & 15;

    // A fragment (16x4 f32): lanes 0-15 hold K=0,1; lanes 16-31 hold K=2,3
    v2f afrag;
    {
        int row = wave * 16 + l15;
        int k0  = lhalf * 2;
        afrag.x = Amat[row * 4 + k0];
        afrag.y = Amat[row * 4 + k0 + 1];
    }

    // lane-half h of vgpr r covers Mrow = wave*16 + 8*h + r  ->  C, k = Mrow&3
    const int C0 = wave * 4 + lhalf * 2;   // d[0..3] -> C0, d[4..7] -> C0+1

    for (int sg = 0; sg < 4; ++sg) {
        float acc0 = 0.0f, acc1 = 0.0f;    // sum over ps=(i,k) of v^2
#pragma unroll
        for (int i = 0; i < 4; ++i) {
            // B fragment (4x16 f32): lanes 0-15 hold K(=j)=0,1; 16-31 hold 2,3
            v2f bfrag;
            int col = i * 64 + sg * 16 + l15;
            int j0  = lhalf * 2;
            bfrag.x = Bmat[j0 * 256 + col];
            bfrag.y = Bmat[(j0 + 1) * 256 + col];
            v8f cz = {0.f, 0.f, 0.f, 0.f, 0.f, 0.f, 0.f, 0.f};
            v8f d = __builtin_amdgcn_wmma_f32_16x16x4_f32(
                false, afrag, false, bfrag, (short)0, cz, false, false);
            acc0 += d[0]*d[0] + d[1]*d[1] + d[2]*d[2] + d[3]*d[3];
            acc1 += d[4]*d[4] + d[5]*d[5] + d[6]*d[6] + d[7]*d[7];
        }
        const int s   = sg * 16 + l15;
        const int pos = b * 64 + s;
        const float n0 = sqrtf(acc0 + EPS_F);
        const float n1 = sqrtf(acc1 + EPS_F);
        atomicMax(&mx[pos * 32 + C0],     __float_as_uint(n0));
        atomicMin(&mn[pos * 32 + C0],     __float_as_uint(n0));
        atomicMax(&mx[pos * 32 + C0 + 1], __float_as_uint(n1));
        atomicMin(&mn[pos * 32 + C0 + 1], __float_as_uint(n1));
    }
}

// ------------------------------------------------------------ dynamic routing
// One block per (b, s): 256 blocks, 256 threads. u recomputed on the fly.
__global__ __launch_bounds__(256)
void routing_kernel(const float* __restrict__ x,
                    const float* __restrict__ W,
                    const unsigned int* __restrict__ mx,
                    const unsigned int* __restrict__ mn,
                    float* __restrict__ out) {
    __shared__ float xp[288 * 16];    // x patch: [B][i*4+j]
    __shared__ float rbuf[288 * 32];  // routing logits
    __shared__ float cbuf[288 * 32];  // softmax(r) over C
    __shared__ float sacc[512];       // sum_B c*u : [C][ps]
    __shared__ float vout[512];       // squashed  : [C][ps]
    __shared__ float scale[32];       // 1/(max-min) per C

    const int tid = threadIdx.x;
    const int pos = blockIdx.x;       // b*64 + s
    const int b   = pos >> 6;
    const int s   = pos & 63;
    const int oh  = s >> 3, ow = s & 7;

    // gather x patch for this position
    for (int idx = tid; idx < 288 * 16; idx += 256) {
        int B = idx >> 4;
        int p = idx & 15;             // p = i*4 + j
        int Borig = B / 9;
        int kk = B % 9;
        int ih = oh * 2 - 1 + kk / 3;
        int iw = ow * 2 - 1 + kk % 3;
        bool ok = (ih >= 0) & (ih < 16) & (iw >= 0) & (iw < 16);
        xp[idx] = ok ? x[(size_t)((b * 32 + Borig) * 16 + p) * 256 + ih * 16 + iw]
                     : 0.0f;
    }
    if (tid < 32) {
        float mxv = __uint_as_float(mx[pos * 32 + tid]);
        float mnv = __uint_as_float(mn[pos * 32 + tid]);
        scale[tid] = 1.0f / (mxv - mnv);
    }
    for (int idx = tid; idx < 288 * 32; idx += 256) rbuf[idx] = 0.0f;
    __syncthreads();

    for (int it = 0; it < 3; ++it) {
        // c = softmax over C (32) per B row
        for (int row = tid; row < 288; row += 256) {
            const float* rr = &rbuf[row * 32];
            float m = rr[0];
            for (int c = 1; c < 32; ++c) m = fmaxf(m, rr[c]);
            float sum = 0.f;
            float* cc = &cbuf[row * 32];
            for (int c = 0; c < 32; ++c) { float e = __expf(rr[c] - m); cc[c] = e; sum += e; }
            float inv = 1.0f / sum;
            for (int c = 0; c < 32; ++c) cc[c] *= inv;
        }
        __syncthreads();

        // sacc[C][ps] = sum_B c[B][C] * u[B][C][ps];  u = scale[C]*(X @ Wm)
#pragma unroll
        for (int half = 0; half < 2; ++half) {
            int cp = tid + half * 256;
            int C  = cp >> 4;
            int ps = cp & 15;
            int i  = ps >> 2;
            int k  = ps & 3;
            float sc  = scale[C];
            float acc = 0.f;
            for (int B = 0; B < 288; ++B) {
                const float* xr = &xp[B * 16 + i * 4];
                const float* wr = &W[(size_t)(B * 32 + C) * 16 + k];
                float u = xr[0]*wr[0] + xr[1]*wr[4] + xr[2]*wr[8] + xr[3]*wr[12];
                acc += cbuf[B * 32 + C] * (sc * u);
            }
            sacc[cp] = acc;
        }
        __syncthreads();

        // squash: v = s / (1 + ||s||)
        if (tid < 32) {
            const float* sr = &sacc[tid * 16];
            float nq = 0.f;
            for (int p = 0; p < 16; ++p) nq += sr[p] * sr[p];
            float f = 1.0f / (1.0f + sqrtf(nq + EPS_F));
            float* vr = &vout[tid * 16];
            for (int p = 0; p < 16; ++p) vr[p] = sr[p] * f;
        }
        __syncthreads();

        if (it < 2) {
            // r[B][C] += sum_ps u[B][C][ps] * v[C][ps]
            for (int idx = tid; idx < 288 * 32; idx += 256) {
                int B = idx >> 5;
                int C = idx & 31;
                const float* xr = &xp[B * 16];
                const float* wr = &W[(size_t)(B * 32 + C) * 16];
                const float* vr = &vout[C * 16];
                float dot = 0.f;
#pragma unroll
                for (int i = 0; i < 4; ++i)
#pragma unroll
                    for (int k = 0; k < 4; ++k) {
                        float u = xr[i*4+0]*wr[0+k] + xr[i*4+1]*wr[4+k]
                                + xr[i*4+2]*wr[8+k] + xr[i*4+3]*wr[12+k];
                        dot += u * vr[i * 4 + k];
                    }
                rbuf[idx] += scale[C] * dot;
            }
            __syncthreads();
        }
    }

    // p_out (4,32,16,8,8)
    for (int idx = tid; idx < 512; idx += 256) {
        int C = idx >> 4;
        int p = idx & 15;
        out[(size_t)((b * 32 + C) * 16 + p) * 64 + s] = vout[idx];
    }
    // a_out (4,32,8,8) appended after 131072 floats
    if (tid < 32) {
        const float* vr = &vout[tid * 16];
        float nq = 0.f;
        for (int p = 0; p < 16; ++p) nq += vr[p] * vr[p];
        out[131072 + (size_t)(b * 32 + tid) * 64 + s] = sqrtf(nq + EPS_F);
    }
}

// -------------------------------------------------------------------- launch
extern "C" void kernel_launch(void* const* d_in, const int* in_sizes, int n_in,
                              void* d_out, int out_size, void* d_ws, size_t ws_size,
                              hipStream_t stream) {
    const float* x = (const float*)d_in[0];
    // d_in[1] = a : patched in the reference but unused by the returned outputs
    const float* W = (const float*)d_in[2];
    float* out = (float*)d_out;

    unsigned int* mx = (unsigned int*)d_ws;       // 4*64*32 = 8192 u32
    unsigned int* mn = mx + 8192;                 // 8192 u32  (64 KB total)

    init_minmax_kernel<<<32, 256, 0, stream>>>(mx, mn);
    vote_norm_kernel<<<4 * 288, 256, 0, stream>>>(x, W, mx, mn);
    routing_kernel<<<256, 256, 0, stream>>>(x, W, mx, mn, out);

    (void)in_sizes; (void)n_in; (void)out_size; (void)ws_size;
}